// DTITAG_34196529611172
// MI455X (gfx1250) — compile-verified
//
#include <hip/hip_runtime.h>
#include <hip/hip_bf16.h>

typedef float v2f __attribute__((ext_vector_type(2)));
typedef float v8f __attribute__((ext_vector_type(8)));

#define N_L 50000
#define E_L 400000
#define N_P 100000
#define E_P 1600000

static __device__ __forceinline__ float frelu(float x) { return fmaxf(x, 0.f); }

// ---------------- generic helpers ----------------
__global__ void k_zero(float* p, int n) {
  int i = blockIdx.x * blockDim.x + threadIdx.x;
  if (i < n) p[i] = 0.f;
}

__global__ void k_pack_feat(const float* __restrict__ in, float* __restrict__ out,
                            int N, int din, int sp) {
  int i = blockIdx.x * blockDim.x + threadIdx.x;
  if (i >= N * sp) return;
  int n = i / sp, f = i % sp;
  out[i] = (f < din) ? in[(size_t)n * din + f] : 0.f;
}

__global__ void k_pack_w(const float* __restrict__ W, float* __restrict__ Wp,
                         int dout, int din, int spo, int spi) {
  int i = blockIdx.x * blockDim.x + threadIdx.x;
  if (i >= spo * spi) return;
  int o = i / spi, c = i % spi;
  Wp[i] = (o < dout && c < din) ? W[(size_t)o * din + c] : 0.f;
}

__global__ void k_pack_b(const float* __restrict__ b, float* __restrict__ bp,
                         int dout, int spo) {
  int i = blockIdx.x * blockDim.x + threadIdx.x;
  if (i < spo) bp[i] = (i < dout) ? b[i] : 0.f;
}

// ---------------- graph: degree + scatter ----------------
__global__ void k_deg(const int* __restrict__ dst, float* __restrict__ deg, int E) {
  int e = blockIdx.x * blockDim.x + threadIdx.x;
  if (e < E) atomicAdd(&deg[dst[e]], 1.f);
}

__global__ void k_invdeg(float* p, int n) {
  int i = blockIdx.x * blockDim.x + threadIdx.x;
  if (i < n) p[i] = 1.f / fmaxf(p[i], 1.f);
}

__global__ void k_scatter(const float* __restrict__ h, const int* __restrict__ src,
                          const int* __restrict__ dst, float* __restrict__ agg,
                          int E, int din, int sp) {
  long long i = (long long)blockIdx.x * blockDim.x + threadIdx.x;
  if (i >= (long long)E * din) return;
  int e = (int)(i / din), f = (int)(i % din);
  atomicAdd(&agg[(size_t)dst[e] * sp + f], h[(size_t)src[e] * sp + f]);
}

// ---------------- fused SAGE layer via V_WMMA_F32_16X16X4_F32 ----------------
// out = relu( h @ Ws^T + (agg * invdeg) @ Wn^T + b ),  all buffers zero-padded.
// A frag (16x4 f32):  lane l -> row M = l%16, k pair base 2*(l/16)
// B frag (4x16 f32):  lane l -> col N = l%16, k pair base 2*(l/16); B[k][n] = W[n][k]
// C/D (16x16 f32):    vgpr v, lane l -> M = v + 8*(l/16), N = l%16
__global__ void k_sage_wmma(const float* __restrict__ h, const float* __restrict__ agg,
                            const float* __restrict__ invdeg,
                            const float* __restrict__ Ws, const float* __restrict__ Wn,
                            const float* __restrict__ bias, float* __restrict__ out,
                            int tilesM, int sin, int sout) {
  int wave = threadIdx.x >> 5;
  int lane = threadIdx.x & 31;
  int tileM = blockIdx.x * (blockDim.x >> 5) + wave;
  if (tileM >= tilesM) return;  // wave-uniform: EXEC stays all-1s for WMMA
  int tileN = blockIdx.y;
  int row = tileM * 16 + (lane & 15);
  int kg = (lane >> 4) << 1;
  int ncol = tileN * 16 + (lane & 15);
  const float* hr = h + (size_t)row * sin;
  const float* ar = agg + (size_t)row * sin;
  const float* wsr = Ws + (size_t)ncol * sin;
  const float* wnr = Wn + (size_t)ncol * sin;
  float idg = invdeg[row];
  v8f acc = {};
  for (int k0 = 0; k0 < sin; k0 += 4) {
    int ka = k0 + kg;
    v2f aS; aS.x = hr[ka];        aS.y = hr[ka + 1];
    v2f bS; bS.x = wsr[ka];       bS.y = wsr[ka + 1];
    acc = __builtin_amdgcn_wmma_f32_16x16x4_f32(false, aS, false, bS, (short)0, acc, false, false);
    v2f aN; aN.x = ar[ka] * idg;  aN.y = ar[ka + 1] * idg;
    v2f bN; bN.x = wnr[ka];       bN.y = wnr[ka + 1];
    acc = __builtin_amdgcn_wmma_f32_16x16x4_f32(false, aN, false, bN, (short)0, acc, false, false);
  }
  float bv = bias[ncol];
  int mb = tileM * 16 + ((lane >> 4) << 3);
#pragma unroll
  for (int v = 0; v < 8; ++v)
    out[(size_t)(mb + v) * sout + ncol] = fmaxf(acc[v] + bv, 0.f);
}

// ---------------- global attention pooling ----------------
__global__ void k_logit(const float* __restrict__ h, int sp, int N, int din,
                        const float* __restrict__ w, const float* __restrict__ b,
                        float* __restrict__ logit, float* __restrict__ partmax) {
  __shared__ float sm[256];
  int i = blockIdx.x * blockDim.x + threadIdx.x;
  float v = -3.4e38f;
  if (i < N) {
    float s = b[0];
    for (int f = 0; f < din; ++f) s += h[(size_t)i * sp + f] * w[f];
    logit[i] = s;
    v = s;
  }
  sm[threadIdx.x] = v;
  __syncthreads();
  for (int st = 128; st > 0; st >>= 1) {
    if (threadIdx.x < st) sm[threadIdx.x] = fmaxf(sm[threadIdx.x], sm[threadIdx.x + st]);
    __syncthreads();
  }
  if (threadIdx.x == 0) partmax[blockIdx.x] = sm[0];
}

__global__ void k_maxred(const float* part, int n, float* outmax) {
  __shared__ float sm[256];
  float v = -3.4e38f;
  for (int i = threadIdx.x; i < n; i += blockDim.x) v = fmaxf(v, part[i]);
  sm[threadIdx.x] = v;
  __syncthreads();
  for (int st = 128; st > 0; st >>= 1) {
    if (threadIdx.x < st) sm[threadIdx.x] = fmaxf(sm[threadIdx.x], sm[threadIdx.x + st]);
    __syncthreads();
  }
  if (threadIdx.x == 0) outmax[0] = sm[0];
}

__global__ void k_gap_accum(const float* __restrict__ h, int sp, int N, int din,
                            const float* __restrict__ logit, const float* __restrict__ maxv,
                            float* __restrict__ acc /* [din+1], acc[din] = Z */) {
  int i = blockIdx.x * blockDim.x + threadIdx.x;
  if (i >= N) return;
  float e = expf(logit[i] - maxv[0]);
  atomicAdd(&acc[din], e);
  for (int f = 0; f < din; ++f) atomicAdd(&acc[f], e * h[(size_t)i * sp + f]);
}

__global__ void k_gap_fin(const float* acc, int din, float* rep) {
  int f = threadIdx.x;
  if (f < din) rep[f] = acc[f] / acc[din];
}

// ---------------- tiny head ----------------
__global__ void k_ttot(const float* __restrict__ rep_l, const float* __restrict__ rep_p,
                       const float* __restrict__ Wt, const float* __restrict__ bt,
                       float* lig3, float* prot3) {
  int j = threadIdx.x;  // 0..63
  if (j >= 64) return;
  const float* rep = (j < 32) ? rep_l : rep_p;
  int o = j & 31;
  float s = bt[o];
  for (int f = 0; f < 31; ++f) s += rep[f] * Wt[o * 31 + f];
  if (j < 32) lig3[o] = s; else prot3[o] = s;
}

// stem conv: implicit input image (1,1,224,32): row0=lig3, row100=prot3, else 0.
// stride 2, SAME (pad lo=0, hi=1 both dims) -> out (32,112,16), fused relu.
__global__ void k_conv1(const float* __restrict__ lig3, const float* __restrict__ prot3,
                        const float* __restrict__ K, const float* __restrict__ B,
                        float* __restrict__ out) {
  int idx = blockIdx.x * blockDim.x + threadIdx.x;
  if (idx >= 32 * 112 * 16) return;
  int ow = idx & 15;
  int oh = (idx >> 4) % 112;
  int c = idx / (112 * 16);
  float s = B[c];
  for (int r = 0; r < 3; ++r) {
    int ih = oh * 2 + r;
    if (ih >= 224) continue;
    const float* row = (ih == 0) ? lig3 : ((ih == 100) ? prot3 : (const float*)0);
    if (!row) continue;
    for (int sx = 0; sx < 3; ++sx) {
      int iw = ow * 2 + sx;
      if (iw >= 32) continue;
      s += K[c * 9 + r * 3 + sx] * row[iw];
    }
  }
  out[idx] = frelu(s);
}

// 32->32 3x3 stride1 SAME (pad 1/1)
__global__ void k_conv3(const float* __restrict__ in, const float* __restrict__ K,
                        const float* __restrict__ B, float* __restrict__ out, int do_relu) {
  int idx = blockIdx.x * blockDim.x + threadIdx.x;
  if (idx >= 32 * 112 * 16) return;
  int x = idx & 15;
  int y = (idx >> 4) % 112;
  int co = idx / (112 * 16);
  float s = B[co];
  for (int ci = 0; ci < 32; ++ci) {
    const float* ip = in + ci * 112 * 16;
    const float* kp = K + (co * 32 + ci) * 9;
    for (int r = 0; r < 3; ++r) {
      int iy = y + r - 1;
      if (iy < 0 || iy >= 112) continue;
      for (int sx = 0; sx < 3; ++sx) {
        int ix = x + sx - 1;
        if (ix < 0 || ix >= 16) continue;
        s += kp[r * 3 + sx] * ip[iy * 16 + ix];
      }
    }
  }
  out[idx] = do_relu ? frelu(s) : s;
}

__global__ void k_res_relu(const float* a, const float* b, float* out) {
  int idx = blockIdx.x * blockDim.x + threadIdx.x;
  if (idx < 32 * 112 * 16) out[idx] = frelu(a[idx] + b[idx]);
}

__global__ void k_chanmean(const float* __restrict__ x, float* __restrict__ xmean) {
  __shared__ float sm[256];
  int c = blockIdx.x;
  float s = 0.f;
  for (int i = threadIdx.x; i < 112 * 16; i += blockDim.x) s += x[c * 112 * 16 + i];
  sm[threadIdx.x] = s;
  __syncthreads();
  for (int st = 128; st > 0; st >>= 1) {
    if (threadIdx.x < st) sm[threadIdx.x] += sm[threadIdx.x + st];
    __syncthreads();
  }
  if (threadIdx.x == 0) xmean[c] = sm[0] / (112.f * 16.f);
}

// 1x1 conv folded with GAP: feat[co] = sum_ci W[co][ci]*xmean[ci] + b[co]
__global__ void k_c2(const float* xmean, const float* __restrict__ W,
                     const float* __restrict__ B, float* feat) {
  int co = blockIdx.x * blockDim.x + threadIdx.x;
  if (co >= 448) return;
  float s = B[co];
  for (int ci = 0; ci < 32; ++ci) s += W[co * 32 + ci] * xmean[ci];
  feat[co] = s;
}

__global__ void k_fc1(const float* feat, const float* __restrict__ W,
                      const float* __restrict__ B, float* o) {
  int j = blockIdx.x * blockDim.x + threadIdx.x;
  if (j >= 4340) return;
  float s = B[j];
  for (int f = 0; f < 448; ++f) s += W[(size_t)j * 448 + f] * feat[f];
  o[j] = frelu(s);
}

__global__ void k_fc2(const float* in, const float* __restrict__ W,
                      const float* __restrict__ B, float* out) {
  __shared__ float sm[256];
  float s = 0.f;
  for (int j = threadIdx.x; j < 4340; j += blockDim.x) s += in[j] * W[j];
  sm[threadIdx.x] = s;
  __syncthreads();
  for (int st = 128; st > 0; st >>= 1) {
    if (threadIdx.x < st) sm[threadIdx.x] += sm[threadIdx.x + st];
    __syncthreads();
  }
  if (threadIdx.x == 0) out[0] = sm[0] + B[0];
}

// ---------------- driver ----------------
extern "C" void kernel_launch(void* const* d_in, const int* in_sizes, int n_in,
                              void* d_out, int out_size, void* d_ws, size_t ws_size,
                              hipStream_t stream) {
  (void)in_sizes; (void)n_in; (void)out_size; (void)ws_size;

  const float* lig_feat  = (const float*)d_in[0];
  const float* prot_feat = (const float*)d_in[1];
  const int*   lig_src   = (const int*)d_in[2];
  const int*   lig_dst   = (const int*)d_in[3];
  const int*   prot_src  = (const int*)d_in[4];
  const int*   prot_dst  = (const int*)d_in[5];
  // lig layer l: d_in[6+3l]=Ws, [7+3l]=Wn, [8+3l]=b ; prot layer l: 21+3l...
  const float* gate_l_w = (const float*)d_in[36];
  const float* gate_l_b = (const float*)d_in[37];
  const float* gate_p_w = (const float*)d_in[38];
  const float* gate_p_b = (const float*)d_in[39];
  const float* ttot_w   = (const float*)d_in[40];
  const float* ttot_b   = (const float*)d_in[41];
  const float* c1_k = (const float*)d_in[42];
  const float* c1_b = (const float*)d_in[43];
  const float* r1_k = (const float*)d_in[44];
  const float* r1_b = (const float*)d_in[45];
  const float* r2_k = (const float*)d_in[46];
  const float* r2_b = (const float*)d_in[47];
  const float* c2_k = (const float*)d_in[48];
  const float* c2_b = (const float*)d_in[49];
  const float* fci_w = (const float*)d_in[50];
  const float* fci_b = (const float*)d_in[51];
  const float* fco_w = (const float*)d_in[52];
  const float* fco_b = (const float*)d_in[53];

  char* wsb = (char*)d_ws;
  size_t off = 0;
  auto alloc = [&](size_t bytes) -> float* {
    float* p = (float*)(wsb + off);
    off = (off + bytes + 255) & ~((size_t)255);
    return p;
  };

  float* hlA  = alloc((size_t)N_L * 80 * 4);
  float* hlB  = alloc((size_t)N_L * 80 * 4);
  float* aggL = alloc((size_t)N_L * 80 * 4);
  float* idL  = alloc((size_t)N_L * 4);
  float* hpA  = alloc((size_t)N_P * 32 * 4);
  float* hpB  = alloc((size_t)N_P * 32 * 4);
  float* aggP = alloc((size_t)N_P * 32 * 4);
  float* idP  = alloc((size_t)N_P * 4);

  const int lig_din[5] = {74, 70, 65, 60, 55};
  const int lig_dout[5] = {70, 65, 60, 55, 31};
  const int lig_spi[5] = {80, 80, 80, 64, 64};
  const int lig_spo[5] = {80, 80, 64, 64, 32};
  float *Wls[5], *Wln[5], *bl[5];
  for (int l = 0; l < 5; ++l) {
    Wls[l] = alloc((size_t)lig_spo[l] * lig_spi[l] * 4);
    Wln[l] = alloc((size_t)lig_spo[l] * lig_spi[l] * 4);
    bl[l]  = alloc((size_t)lig_spo[l] * 4);
  }
  float *Wps[5], *Wpn[5], *bp[5];
  for (int l = 0; l < 5; ++l) {
    Wps[l] = alloc(32 * 32 * 4);
    Wpn[l] = alloc(32 * 32 * 4);
    bp[l]  = alloc(32 * 4);
  }

  float* logits  = alloc((size_t)N_P * 4);
  float* partmax = alloc(1024 * 4);
  float* maxv    = alloc(4);
  float* gapacc  = alloc(32 * 4);
  float* rep_l   = alloc(32 * 4);
  float* rep_p   = alloc(32 * 4);
  float* lig3    = alloc(32 * 4);
  float* prot3   = alloc(32 * 4);
  float* x1      = alloc(32 * 112 * 16 * 4);
  float* hb1     = alloc(32 * 112 * 16 * 4);
  float* hb2     = alloc(32 * 112 * 16 * 4);
  float* xmean   = alloc(32 * 4);
  float* feat    = alloc(448 * 4);
  float* fc1     = alloc(4340 * 4);

  auto G = [](long long n) { return (unsigned)((n + 255) / 256); };

  // --- pack features + weights into zero-padded buffers ---
  k_pack_feat<<<G((long long)N_L * 80), 256, 0, stream>>>(lig_feat, hlA, N_L, 74, 80);
  k_pack_feat<<<G((long long)N_P * 32), 256, 0, stream>>>(prot_feat, hpA, N_P, 31, 32);
  for (int l = 0; l < 5; ++l) {
    const float* Ws = (const float*)d_in[6 + 3 * l];
    const float* Wn = (const float*)d_in[7 + 3 * l];
    const float* b  = (const float*)d_in[8 + 3 * l];
    int spo = lig_spo[l], spi = lig_spi[l];
    k_pack_w<<<G((long long)spo * spi), 256, 0, stream>>>(Ws, Wls[l], lig_dout[l], lig_din[l], spo, spi);
    k_pack_w<<<G((long long)spo * spi), 256, 0, stream>>>(Wn, Wln[l], lig_dout[l], lig_din[l], spo, spi);
    k_pack_b<<<1, 256, 0, stream>>>(b, bl[l], lig_dout[l], spo);
  }
  for (int l = 0; l < 5; ++l) {
    const float* Ws = (const float*)d_in[21 + 3 * l];
    const float* Wn = (const float*)d_in[22 + 3 * l];
    const float* b  = (const float*)d_in[23 + 3 * l];
    k_pack_w<<<G(32 * 32), 256, 0, stream>>>(Ws, Wps[l], 31, 31, 32, 32);
    k_pack_w<<<G(32 * 32), 256, 0, stream>>>(Wn, Wpn[l], 31, 31, 32, 32);
    k_pack_b<<<1, 256, 0, stream>>>(b, bp[l], 31, 32);
  }

  // --- inverse degree (once per graph) ---
  k_zero<<<G(N_L), 256, 0, stream>>>(idL, N_L);
  k_deg<<<G(E_L), 256, 0, stream>>>(lig_dst, idL, E_L);
  k_invdeg<<<G(N_L), 256, 0, stream>>>(idL, N_L);
  k_zero<<<G(N_P), 256, 0, stream>>>(idP, N_P);
  k_deg<<<G(E_P), 256, 0, stream>>>(prot_dst, idP, E_P);
  k_invdeg<<<G(N_P), 256, 0, stream>>>(idP, N_P);

  // --- ligand GNN (5 fused SAGE layers, WMMA) ---
  float* cur = hlA; float* nxt = hlB;
  for (int l = 0; l < 5; ++l) {
    int spi = lig_spi[l], spo = lig_spo[l], din = lig_din[l];
    k_zero<<<G((long long)N_L * spi), 256, 0, stream>>>(aggL, N_L * spi);
    k_scatter<<<G((long long)E_L * din), 256, 0, stream>>>(cur, lig_src, lig_dst, aggL, E_L, din, spi);
    dim3 grd((N_L / 16 + 7) / 8, spo / 16);
    k_sage_wmma<<<grd, 256, 0, stream>>>(cur, aggL, idL, Wls[l], Wln[l], bl[l], nxt,
                                         N_L / 16, spi, spo);
    float* t = cur; cur = nxt; nxt = t;
  }
  float* h_l_final = cur;  // stride 32

  // --- protein GNN ---
  cur = hpA; nxt = hpB;
  for (int l = 0; l < 5; ++l) {
    k_zero<<<G((long long)N_P * 32), 256, 0, stream>>>(aggP, N_P * 32);
    k_scatter<<<G((long long)E_P * 31), 256, 0, stream>>>(cur, prot_src, prot_dst, aggP, E_P, 31, 32);
    dim3 grd((N_P / 16 + 7) / 8, 2);
    k_sage_wmma<<<grd, 256, 0, stream>>>(cur, aggP, idP, Wps[l], Wpn[l], bp[l], nxt,
                                         N_P / 16, 32, 32);
    float* t = cur; cur = nxt; nxt = t;
  }
  float* h_p_final = cur;  // stride 32

  // --- global attention pooling ---
  auto gap = [&](const float* h, int N, const float* w, const float* b, float* rep) {
    unsigned nb = G(N);
    k_logit<<<nb, 256, 0, stream>>>(h, 32, N, 31, w, b, logits, partmax);
    k_maxred<<<1, 256, 0, stream>>>(partmax, (int)nb, maxv);
    k_zero<<<1, 256, 0, stream>>>(gapacc, 32);
    k_gap_accum<<<nb, 256, 0, stream>>>(h, 32, N, 31, logits, maxv, gapacc);
    k_gap_fin<<<1, 32, 0, stream>>>(gapacc, 31, rep);
  };
  gap(h_l_final, N_L, gate_l_w, gate_l_b, rep_l);
  gap(h_p_final, N_P, gate_p_w, gate_p_b, rep_p);

  // --- head ---
  k_ttot<<<1, 64, 0, stream>>>(rep_l, rep_p, ttot_w, ttot_b, lig3, prot3);
  k_conv1<<<G(32 * 112 * 16), 256, 0, stream>>>(lig3, prot3, c1_k, c1_b, x1);
  k_conv3<<<G(32 * 112 * 16), 256, 0, stream>>>(x1, r1_k, r1_b, hb1, 1);
  k_conv3<<<G(32 * 112 * 16), 256, 0, stream>>>(hb1, r2_k, r2_b, hb2, 0);
  k_res_relu<<<G(32 * 112 * 16), 256, 0, stream>>>(x1, hb2, hb1);
  k_chanmean<<<32, 256, 0, stream>>>(hb1, xmean);
  k_c2<<<2, 256, 0, stream>>>(xmean, c2_k, c2_b, feat);
  k_fc1<<<G(4340), 256, 0, stream>>>(feat, fci_w, fci_b, fc1);
  k_fc2<<<1, 256, 0, stream>>>(fc1, fco_w, fco_b, (float*)d_out);
}